// L1OutUB_9216999817952
// MI455X (gfx1250) — compile-verified
//
#include <hip/hip_runtime.h>
#include <math.h>

// ---------------------------------------------------------------------------
// CMI leave-one-out estimator as ONE fp32 WMMA GEMM:
//   hidden = Wbig^T (1024x64) @ x^T (64 x B)
// Wbig column group g<63 = zero-row-inserted LOO weights for n=g
// (8 cols mun-padded + 8 cols lvn-padded); group 63 = pmu/plv (positive).
// Transposed D-tile layout puts all 8 hidden units of one (batch, n) pair
// in-lane; one shfl_xor(16) pairs mun<->lvn for the NLL epilogue.
//
// v2: each wave processes TWO 16-batch blocks per A load (halves L2 weight
// traffic, two independent WMMA accumulator chains for ILP) and software-
// pipelines the A stream with a ping-pong register double buffer.
// ---------------------------------------------------------------------------

typedef __attribute__((ext_vector_type(2))) float v2f;
typedef __attribute__((ext_vector_type(8))) float v8f;

#define B_TOTAL 32768
#define D_FEAT  64
#define K_OUT   8
#define NGROUPS 64            // 63 LOO groups + 1 positive group
#define KSTEPS  16            // 64 / 4 (K per WMMA = 4)
#define UB      2             // batch blocks per wave
#define NWAVES  (B_TOTAL/(16*UB))           // 1024 waves
#define APACK_FLOATS (NGROUPS*KSTEPS*32*2)  // 65536 floats = 256 KiB

// Build the WMMA-A-layout packed weight stream.
// apack[ ((g*16 + t)*32 + L)*2 + v ] = Wbig[k, c]  with
//   c = g*16 + (L&15),  k = 4t + ((L>>4)<<1) + v
__global__ void build_apack(const float* __restrict__ pmu_W1,
                            const float* __restrict__ plv_W1,
                            const float* __restrict__ pmun_W1,
                            const float* __restrict__ plvn_W1,
                            float* __restrict__ apack) {
  int id = blockIdx.x * blockDim.x + threadIdx.x;
  if (id >= APACK_FLOATS) return;
  int v = id & 1;
  int L = (id >> 1) & 31;
  int t = (id >> 6) & 15;
  int g = id >> 10;
  int u = L & 15;                       // column within 16-wide group
  int k = 4 * t + ((L >> 4) << 1) + v;  // feature row, 0..63
  int h = u & 7;                        // hidden unit (7 == zero pad)
  float val = 0.0f;
  if (h < 7) {
    if (g == 63) {                      // positive path: pmu (u<8) / plv
      const float* src = (u < 8) ? pmu_W1 : plv_W1;   // (64,7) row-major
      val = src[k * 7 + h];
    } else {                            // LOO group n=g: pmun / plvn
      int n = g;
      if (k != n) {
        int kk = (k < n) ? k : (k - 1);
        const float* src = (u < 8) ? pmun_W1 : plvn_W1; // (63,7) row-major
        val = src[kk * 7 + h];
      }
    }
  }
  apack[id] = val;
}

__global__ __launch_bounds__(256) void cmi_main(
    const float* __restrict__ x, const float* __restrict__ y,
    const float* __restrict__ pmu_b1,  const float* __restrict__ pmu_W2,  const float* __restrict__ pmu_b2,
    const float* __restrict__ plv_b1,  const float* __restrict__ plv_W2,  const float* __restrict__ plv_b2,
    const float* __restrict__ pmun_b1, const float* __restrict__ pmun_W2, const float* __restrict__ pmun_b2,
    const float* __restrict__ plvn_b1, const float* __restrict__ plvn_W2, const float* __restrict__ plvn_b2,
    const float* __restrict__ apack, float* __restrict__ partial) {
  const int lane = threadIdx.x & 31;
  const int wave = blockIdx.x * (blockDim.x >> 5) + (threadIdx.x >> 5);
  const int b0   = wave * 16 * UB;  // this wave's batches: UB 16-row blocks
  const int half = lane >> 4;       // 0 = mu/mun side, 1 = lv/lvn side
  const int col  = lane & 15;       // batch within block (N of the D tile)

  // per-half layer-2 / bias params (tiny, cached)
  const float* nb1 = half ? plvn_b1 : pmun_b1;
  const float* nw2 = half ? plvn_W2 : pmun_W2;
  const float  nb2 = half ? plvn_b2[0] : pmun_b2[0];
  const float* pb1 = half ? plv_b1 : pmu_b1;
  const float* pw2 = half ? plv_W2 : pmu_W2;
  const float  pb2 = half ? plv_b2[0] : pmu_b2[0];
  float nb1r[7], nw2r[7], pb1r[7], pw2r[7];
  #pragma unroll
  for (int r = 0; r < 7; ++r) {
    nb1r[r] = nb1[r]; nw2r[r] = nw2[r]; pb1r[r] = pb1[r]; pw2r[r] = pw2[r];
  }

  // Preload B-operand (x^T) tiles once; reused across all 64 M-groups.
  // B lane layout: lane = N-col (batch), VGPR v = K = 4t + (lane>>4)*2 + v.
  v2f bx[UB][KSTEPS];
  #pragma unroll
  for (int u = 0; u < UB; ++u) {
    const float* xrow = x + (size_t)(b0 + u * 16 + col) * D_FEAT + (half << 1);
    #pragma unroll
    for (int t = 0; t < KSTEPS; ++t)
      bx[u][t] = *(const v2f*)(xrow + 4 * t);
  }

  float negA[UB], negB[UB], negC[UB], posA[UB], posB[UB], posC[UB];
  #pragma unroll
  for (int u = 0; u < UB; ++u) {
    negA[u] = negB[u] = negC[u] = 0.f;
    posA[u] = posB[u] = posC[u] = 0.f;
  }

  const v2f* ap = (const v2f*)apack;

  auto loadA = [&](v2f* dst, int g) {
    #pragma unroll
    for (int t = 0; t < KSTEPS; ++t)
      dst[t] = ap[(g * 16 + t) * 32 + lane];  // coalesced 8B/lane
  };

  auto computeG = [&](int g, const v2f* a) {
    v8f acc[UB];
    #pragma unroll
    for (int u = 0; u < UB; ++u) acc[u] = v8f{0.f,0.f,0.f,0.f,0.f,0.f,0.f,0.f};
    // two independent accumulator chains share each A tile
    #pragma unroll
    for (int t = 0; t < KSTEPS; ++t) {
      #pragma unroll
      for (int u = 0; u < UB; ++u) {
        acc[u] = __builtin_amdgcn_wmma_f32_16x16x4_f32(
            /*neg_a=*/false, a[t], /*neg_b=*/false, bx[u][t],
            /*c_mod=*/(short)0, acc[u], /*reuse_a=*/false, /*reuse_b=*/false);
      }
    }
    const bool pos = (g == NGROUPS - 1);
    #pragma unroll
    for (int u = 0; u < UB; ++u) {
      float val = pos ? pb2 : nb2;
      #pragma unroll
      for (int r = 0; r < 7; ++r) {
        float hr = acc[u][r] + (pos ? pb1r[r] : nb1r[r]);
        hr = hr > 0.f ? hr : 0.f;                 // relu
        val += hr * (pos ? pw2r[r] : nw2r[r]);    // layer 2
      }
      // pair mun (low half) with lvn (high half) for the same (b, n)
      float other = __shfl_xor(val, 16, 32);
      if (lane < 16) {
        float mval = val;                 // mu / mun
        float lvt  = tanhf(other);        // lv / lvn
        float c0 = 0.5f * __expf(-lvt);   // 1/(2 exp(lv))
        float a_ = -c0 * mval * mval - 0.5f * lvt;
        float b_ = 2.f * c0 * mval;
        float c_ = -c0;
        if (pos) { posA[u] = a_; posB[u] = b_; posC[u] = c_; }
        else     { negA[u] += a_; negB[u] += b_; negC[u] += c_; }
      }
    }
  };

  // Software pipeline: ping-pong A buffers, prefetch group g+1 / g+2
  // while the WMMA chains + epilogue for the current group execute.
  v2f a0[KSTEPS], a1[KSTEPS];
  loadA(a0, 0);
  for (int g = 0; g < NGROUPS; g += 2) {
    loadA(a1, g + 1);                       // NGROUPS even: g+1 <= 63
    computeG(g, a0);
    if (g + 2 < NGROUPS) loadA(a0, g + 2);
    computeG(g + 1, a1);
  }

  // per-k contribution:  pos/B - neg/(B*63), expanded in powers of y[b,k]
  const float invB  = 1.0f / (float)B_TOTAL;
  const float invBM = invB / 63.0f;
  float tk[K_OUT];
  if (lane < 16) {
    #pragma unroll
    for (int k = 0; k < K_OUT; ++k) tk[k] = 0.f;
    #pragma unroll
    for (int u = 0; u < UB; ++u) {
      const float* yb = y + (size_t)(b0 + u * 16 + col) * K_OUT;
      #pragma unroll
      for (int k = 0; k < K_OUT; ++k) {
        float yv = yb[k];
        float p = posA[u] + posB[u] * yv + posC[u] * yv * yv;
        float n = negA[u] + negB[u] * yv + negC[u] * yv * yv;
        tk[k] += p * invB - n * invBM;
      }
    }
  } else {
    #pragma unroll
    for (int k = 0; k < K_OUT; ++k) tk[k] = 0.f;
  }
  // deterministic in-wave tree reduction, one partial row per wave
  #pragma unroll
  for (int k = 0; k < K_OUT; ++k) {
    float s = tk[k];
    s += __shfl_xor(s, 1, 32);
    s += __shfl_xor(s, 2, 32);
    s += __shfl_xor(s, 4, 32);
    s += __shfl_xor(s, 8, 32);
    s += __shfl_xor(s, 16, 32);
    if (lane == 0) partial[wave * K_OUT + k] = s;
  }
}

// Deterministic final reduction over the wave partials.
__global__ void cmi_reduce(const float* __restrict__ partial,
                           float* __restrict__ out) {
  int k = threadIdx.x;
  if (k < K_OUT) {
    float s = 0.f;
    for (int w = 0; w < NWAVES; ++w) s += partial[w * K_OUT + k];
    out[k] = fabsf(s);
  }
}

extern "C" void kernel_launch(void* const* d_in, const int* in_sizes, int n_in,
                              void* d_out, int out_size, void* d_ws, size_t ws_size,
                              hipStream_t stream) {
  const float* x        = (const float*)d_in[0];
  const float* y        = (const float*)d_in[1];
  const float* pmu_W1   = (const float*)d_in[2];
  const float* pmu_b1   = (const float*)d_in[3];
  const float* pmu_W2   = (const float*)d_in[4];
  const float* pmu_b2   = (const float*)d_in[5];
  const float* plv_W1   = (const float*)d_in[6];
  const float* plv_b1   = (const float*)d_in[7];
  const float* plv_W2   = (const float*)d_in[8];
  const float* plv_b2   = (const float*)d_in[9];
  const float* pmun_W1  = (const float*)d_in[10];
  const float* pmun_b1  = (const float*)d_in[11];
  const float* pmun_W2  = (const float*)d_in[12];
  const float* pmun_b2  = (const float*)d_in[13];
  const float* plvn_W1  = (const float*)d_in[14];
  const float* plvn_b1  = (const float*)d_in[15];
  const float* plvn_W2  = (const float*)d_in[16];
  const float* plvn_b2  = (const float*)d_in[17];

  float* apack   = (float*)d_ws;                 // 256 KiB
  float* partial = apack + APACK_FLOATS;         // 1024*8 floats = 32 KiB

  build_apack<<<APACK_FLOATS / 256, 256, 0, stream>>>(
      pmu_W1, plv_W1, pmun_W1, plvn_W1, apack);

  // 1024 waves, 8 waves (256 threads) per block -> 128 blocks
  cmi_main<<<NWAVES / 8, 256, 0, stream>>>(
      x, y,
      pmu_b1, pmu_W2, pmu_b2, plv_b1, plv_W2, plv_b2,
      pmun_b1, pmun_W2, pmun_b2, plvn_b1, plvn_W2, plvn_b2,
      apack, partial);

  cmi_reduce<<<1, 32, 0, stream>>>(partial, (float*)d_out);
}